// SAGE_54674933678409
// MI455X (gfx1250) — compile-verified
//
#include <hip/hip_runtime.h>

#define N_NODES 100000
#define N_EDGES 1600000
#define DIM 128
#define LDS_STRIDE 132   // 128 + 4 pad: avoids 64-bank conflicts on row-major reads

typedef __attribute__((ext_vector_type(2))) float v2f;
typedef __attribute__((ext_vector_type(8))) float v8f;

// ---------------- degree kernels ----------------
__global__ void SAGE_deg_kernel(const int* __restrict__ dst,
                                float* __restrict__ deg, int n_edges) {
    int e = blockIdx.x * blockDim.x + threadIdx.x;
    if (e < n_edges) atomicAdd(&deg[dst[e]], 1.0f);
}

__global__ void SAGE_invdeg_kernel(float* __restrict__ deg, int n) {
    int i = blockIdx.x * blockDim.x + threadIdx.x;
    if (i < n) deg[i] = 1.0f / fmaxf(deg[i], 1.0f);
}

// ---------------- edge scatter: neigh[dst] += x[src] ----------------
// One wave32 per edge; each lane moves 4 contiguous floats (float4 gather,
// 4x global_atomic_add_f32 scatter). Both buffers are L2-resident.
__global__ void SAGE_scatter_kernel(const float* __restrict__ x,
                                    const int* __restrict__ src,
                                    const int* __restrict__ dst,
                                    float* __restrict__ neigh, int n_edges) {
    int t = blockIdx.x * blockDim.x + threadIdx.x;
    int e = t >> 5;
    int lane = t & 31;
    if (e >= n_edges) return;
    int s = src[e];
    int d = dst[e];
    float4 v = ((const float4*)(x + (size_t)s * DIM))[lane];
    float* op = neigh + (size_t)d * DIM + (size_t)lane * 4;
    atomicAdd(op + 0, v.x);
    atomicAdd(op + 1, v.y);
    atomicAdd(op + 2, v.z);
    atomicAdd(op + 3, v.w);
}

// ---------------- fused dual-GEMM + bias + relu via fp32 WMMA ----------------
// Block = 16 output rows x 128 cols. 8 waves, each owns a 16x16 column tile.
// acc = x_tile @ Ws + (neigh_tile * invdeg) @ Wn, K=128 in steps of 4.
__global__ void __launch_bounds__(256)
SAGE_gemm_kernel(const float* __restrict__ x, const float* __restrict__ neigh,
                 const float* __restrict__ invdeg,
                 const float* __restrict__ Ws, const float* __restrict__ Wn,
                 const float* __restrict__ bias, float* __restrict__ out,
                 int relu) {
    __shared__ float xs[16 * LDS_STRIDE];
    __shared__ float ns[16 * LDS_STRIDE];

    const int row0 = blockIdx.x * 16;
    const int tid = threadIdx.x;

    // Cooperative stage of 16x128 x-tile and invdeg-scaled neigh-tile into LDS.
    for (int i = tid; i < 16 * 32; i += 256) {
        int r = i >> 5;
        int c4 = (i & 31) * 4;
        size_t goff = (size_t)(row0 + r) * DIM + c4;
        float4 vx = *(const float4*)(x + goff);
        float4 vn = *(const float4*)(neigh + goff);
        float id = invdeg[row0 + r];
        vn.x *= id; vn.y *= id; vn.z *= id; vn.w *= id;
        *(float4*)(&xs[r * LDS_STRIDE + c4]) = vx;
        *(float4*)(&ns[r * LDS_STRIDE + c4]) = vn;
    }
    __syncthreads();

    const int wave = tid >> 5;     // column tile 0..7
    const int lane = tid & 31;
    const int half = lane >> 4;    // lane half selects K pair / row offset
    const int l16  = lane & 15;
    const int colg = wave * 16 + l16;

    const float* arow_x = &xs[l16 * LDS_STRIDE];
    const float* arow_n = &ns[l16 * LDS_STRIDE];

    v8f acc = {};
    #pragma unroll
    for (int k = 0; k < DIM; k += 4) {
        const int k0 = k + half * 2;           // this lane's K pair
        // A frag: lane = M row, VGPRs hold K=k0,k0+1 (fp32 16x4 layout)
        v2f ax = *(const v2f*)(arow_x + k0);
        // B frag: lane = N col, VGPRs hold K=k0,k0+1 (fp32 4x16 layout)
        v2f bx;
        bx.x = Ws[(size_t)k0 * DIM + colg];
        bx.y = Ws[(size_t)(k0 + 1) * DIM + colg];
        acc = __builtin_amdgcn_wmma_f32_16x16x4_f32(false, ax, false, bx,
                                                    (short)0, acc, false, false);
        v2f an = *(const v2f*)(arow_n + k0);
        v2f bn;
        bn.x = Wn[(size_t)k0 * DIM + colg];
        bn.y = Wn[(size_t)(k0 + 1) * DIM + colg];
        acc = __builtin_amdgcn_wmma_f32_16x16x4_f32(false, an, false, bn,
                                                    (short)0, acc, false, false);
    }

    const float bv = bias[colg];
    #pragma unroll
    for (int r = 0; r < 8; r++) {
        // C/D layout: VGPR r -> row r (lanes 0-15) / row r+8 (lanes 16-31)
        float v = acc[r] + bv;
        if (relu) v = fmaxf(v, 0.0f);
        out[(size_t)(row0 + r + half * 8) * DIM + colg] = v;
    }
}

extern "C" void kernel_launch(void* const* d_in, const int* in_sizes, int n_in,
                              void* d_out, int out_size, void* d_ws, size_t ws_size,
                              hipStream_t stream) {
    const float* emb    = (const float*)d_in[0];  // [N,128]
    const float* Wself  = (const float*)d_in[1];  // [3,128,128]
    const float* Wneigh = (const float*)d_in[2];  // [3,128,128]
    const float* bias   = (const float*)d_in[3];  // [3,128]
    const int*   src    = (const int*)d_in[4];    // [E]
    const int*   dst    = (const int*)d_in[5];    // [E]
    float* out = (float*)d_out;

    float* invdeg = (float*)d_ws;                 // N floats
    float* neigh  = invdeg + N_NODES;             // N*128 floats

    // in-degree -> 1/max(deg,1)
    hipMemsetAsync(invdeg, 0, (size_t)N_NODES * sizeof(float), stream);
    SAGE_deg_kernel<<<(N_EDGES + 255) / 256, 256, 0, stream>>>(dst, invdeg, N_EDGES);
    SAGE_invdeg_kernel<<<(N_NODES + 255) / 256, 256, 0, stream>>>(invdeg, N_NODES);

    const float* xin = emb;
    for (int l = 0; l < 3; l++) {
        hipMemsetAsync(neigh, 0, (size_t)N_NODES * DIM * sizeof(float), stream);
        SAGE_scatter_kernel<<<(N_EDGES * 32 + 255) / 256, 256, 0, stream>>>(
            xin, src, dst, neigh, N_EDGES);
        SAGE_gemm_kernel<<<N_NODES / 16, 256, 0, stream>>>(
            xin, neigh, invdeg,
            Wself + (size_t)l * DIM * DIM,
            Wneigh + (size_t)l * DIM * DIM,
            bias + (size_t)l * DIM,
            out, (l < 2) ? 1 : 0);
        xin = out;  // layers 1,2 run in-place on d_out (tile-local read-then-write)
    }
}